// LorenzOscillator_81509889344059
// MI455X (gfx1250) — compile-verified
//
#include <hip/hip_runtime.h>
#include <hip/hip_bf16.h>

#define L_SIGMA 10.0f
#define L_RHO   28.0f
#define L_BETA  (8.0f / 3.0f)
#define L_H     0.01f
#define L_NRK   10
#define IN_DIM  80
#define HID     512
#define SEQ     512

typedef __attribute__((ext_vector_type(2))) float v2f;
typedef __attribute__((ext_vector_type(8))) float v8f;

// ---------------------------------------------------------------------------
// Kernel A: precompute C[t] = W_c @ (x[:,t,:] @ W_in.T + b_in)  for all t.
// One thread per (t, j) column; writes C[t][i][j] (row-major [t][3][3]).
// ---------------------------------------------------------------------------
__global__ void lorenz_precompute_c(const float* __restrict__ x,
                                    const float* __restrict__ W_in,
                                    const float* __restrict__ b_in,
                                    const float* __restrict__ W_c,
                                    float* __restrict__ Cws) {
    int idx = blockIdx.x * blockDim.x + threadIdx.x;
    if (idx >= SEQ * 3) return;
    int t = idx / 3;
    int j = idx % 3;

    const float* wrow = W_in + j * IN_DIM;
    float u[3];
#pragma unroll
    for (int m = 0; m < 3; ++m) {
        const float* xr = x + m * (SEQ * IN_DIM) + t * IN_DIM;
        float acc = 0.0f;
#pragma unroll 8
        for (int k = 0; k < IN_DIM; ++k) acc = fmaf(xr[k], wrow[k], acc);
        u[m] = acc + b_in[j];
    }
#pragma unroll
    for (int i = 0; i < 3; ++i) {
        Cws[t * 9 + i * 3 + j] =
            W_c[i * 3 + 0] * u[0] + W_c[i * 3 + 1] * u[1] + W_c[i * 3 + 2] * u[2];
    }
}

// ---------------------------------------------------------------------------
// deriv with the reference's stacking quirk:
//   per column j: dx,dy,dz computed from rows of m; results stored as
//   k[j][0]=dx_j, k[j][1]=dy_j, k[j][2]=dz_j  (stack along dim 1).
// ---------------------------------------------------------------------------
__device__ __forceinline__ void lorenz_deriv(const float* m, const float* c, float* k) {
#pragma unroll
    for (int j = 0; j < 3; ++j) {
        float xx = m[0 + j], yy = m[3 + j], zz = m[6 + j];
        float dx = L_SIGMA * (yy - xx) + c[0 + j];
        float dy = xx * (L_RHO - zz) - yy + c[3 + j];
        float dz = xx * yy - L_BETA * zz + c[6 + j];
        k[j * 3 + 0] = dx;
        k[j * 3 + 1] = dy;
        k[j * 3 + 2] = dz;
    }
}

// ---------------------------------------------------------------------------
// Kernel B: the inherently sequential scan. One lane carries the 9-float
// state through 512 steps x 10 RK4 substeps; stores S[t][b][j] after each t.
// ---------------------------------------------------------------------------
__global__ void lorenz_scan(const float* __restrict__ Cws, float* __restrict__ Sws) {
    if (threadIdx.x != 0) return;

    float s[9];
#pragma unroll
    for (int i = 0; i < 9; ++i) s[i] = 0.0f;

#pragma unroll 1
    for (int t = 0; t < SEQ; ++t) {
        float c[9];
#pragma unroll
        for (int i = 0; i < 9; ++i) c[i] = Cws[t * 9 + i];

#pragma unroll 1
        for (int r = 0; r < L_NRK; ++r) {
            float k1[9], k2[9], k3[9], k4[9], tmp[9];
            lorenz_deriv(s, c, k1);
#pragma unroll
            for (int i = 0; i < 9; ++i) tmp[i] = fmaf(0.5f * L_H, k1[i], s[i]);
            lorenz_deriv(tmp, c, k2);
#pragma unroll
            for (int i = 0; i < 9; ++i) tmp[i] = fmaf(0.5f * L_H, k2[i], s[i]);
            lorenz_deriv(tmp, c, k3);
#pragma unroll
            for (int i = 0; i < 9; ++i) tmp[i] = fmaf(L_H, k3[i], s[i]);
            lorenz_deriv(tmp, c, k4);
#pragma unroll
            for (int i = 0; i < 9; ++i)
                s[i] = fmaf(L_H / 6.0f,
                            k1[i] + 2.0f * k2[i] + 2.0f * k3[i] + k4[i], s[i]);
        }
#pragma unroll
        for (int i = 0; i < 9; ++i) Sws[t * 9 + i] = s[i];
    }
}

// ---------------------------------------------------------------------------
// Kernel C: out[b][t][h] = sum_j S[t][b][j] * W_out[h][j] + b_out[h]
// as a [1536 x 3] x [3 x 512] GEMM with K padded to 4, using
// V_WMMA_F32_16X16X4_F32. One wave per 16x16 output tile.
//
// A layout (16x4 f32): lanes 0-15 hold M=lane, VGPR0=K0, VGPR1=K1;
//                      lanes 16-31 hold M=lane-16, VGPR0=K2, VGPR1=K3(=0).
// B layout (4x16 f32): mirrored, lane%16 = N.
// C/D layout: VGPR v holds row M=v (lanes 0-15) / M=v+8 (lanes 16-31), N=lane%16.
// ---------------------------------------------------------------------------
__global__ void lorenz_out_wmma(const float* __restrict__ Sws,
                                const float* __restrict__ W_out,
                                const float* __restrict__ b_out,
                                float* __restrict__ out) {
    int tile_m = blockIdx.x;           // 0..95  (rows r = b*512 + t)
    int tile_n = blockIdx.y;           // 0..31  (hidden columns)
    int lane   = threadIdx.x;          // 0..31, one wave per block
    int lo     = lane & 15;
    int hi     = lane >> 4;            // 0: K=0,1  1: K=2,3

    // --- A fragment: rows of the state matrix, K in {0,1,2}, K=3 padded 0 ---
    int rg = tile_m * 16 + lo;         // global output row
    int t  = rg & (SEQ - 1);
    int b  = rg >> 9;
    const float* srow = Sws + t * 9 + b * 3;
    v2f a;
    a.x = hi ? srow[2] : srow[0];
    a.y = hi ? 0.0f    : srow[1];

    // --- B fragment: W_out[h][k] transposed into 4x16 (K x N) ---
    int ng = tile_n * 16 + lo;         // global hidden column
    const float* wrow = W_out + ng * 3;
    v2f bm;
    bm.x = hi ? wrow[2] : wrow[0];
    bm.y = hi ? 0.0f    : wrow[1];

    // --- C init with bias broadcast: C[m][n] = b_out[n] for all m ---
    float bias = b_out[ng];
    v8f acc;
#pragma unroll
    for (int v = 0; v < 8; ++v) acc[v] = bias;

    // D = A x B + C   (v_wmma_f32_16x16x4_f32)
    acc = __builtin_amdgcn_wmma_f32_16x16x4_f32(
        /*neg_a=*/false, a, /*neg_b=*/false, bm,
        /*c_mod=*/(short)0, acc, /*reuse_a=*/false, /*reuse_b=*/false);

    // --- scatter store per C/D layout ---
#pragma unroll
    for (int v = 0; v < 8; ++v) {
        int M  = v + (hi ? 8 : 0);
        int r  = tile_m * 16 + M;
        int tb = r >> 9;
        int tt = r & (SEQ - 1);
        out[(size_t)tb * (SEQ * HID) + (size_t)tt * HID + ng] = acc[v];
    }
}

// ---------------------------------------------------------------------------
extern "C" void kernel_launch(void* const* d_in, const int* in_sizes, int n_in,
                              void* d_out, int out_size, void* d_ws, size_t ws_size,
                              hipStream_t stream) {
    const float* x     = (const float*)d_in[0];
    const float* W_in  = (const float*)d_in[1];
    const float* b_in  = (const float*)d_in[2];
    const float* W_c   = (const float*)d_in[3];
    const float* W_out = (const float*)d_in[4];
    const float* b_out = (const float*)d_in[5];
    float* out = (float*)d_out;

    float* Cws = (float*)d_ws;          // [SEQ][3][3]
    float* Sws = Cws + SEQ * 9;         // [SEQ][3][3]

    lorenz_precompute_c<<<dim3((SEQ * 3 + 255) / 256), 256, 0, stream>>>(
        x, W_in, b_in, W_c, Cws);
    lorenz_scan<<<1, 32, 0, stream>>>(Cws, Sws);
    lorenz_out_wmma<<<dim3(96, 32), 32, 0, stream>>>(Sws, W_out, b_out, out);
}